// GraphAttentionLayer_32220844655111
// MI455X (gfx1250) — compile-verified
//
#include <hip/hip_runtime.h>
#include <hip/hip_bf16.h>

// Problem constants (match reference)
#define NN    50000      // nodes
#define NE    600000     // edges
#define NDIM  128        // node feature dim
#define EDIM  64         // edge feature dim
#define NH    4          // heads
#define HD    32         // head dim  (NH*HD == 128)
#define ODIM  128        // output dim per node

typedef __attribute__((ext_vector_type(2))) float v2f;
typedef __attribute__((ext_vector_type(8))) float v8f;

// ---------------------------------------------------------------------------
// Kernel Z: zero output, zero mraw (m>=0 by construction), zero deg
// ---------------------------------------------------------------------------
__global__ void k_init(float* __restrict__ out, float* __restrict__ mraw,
                       unsigned* __restrict__ deg) {
    int i = blockIdx.x * blockDim.x + threadIdx.x;
    if (i < NN * ODIM) out[i] = 0.0f;
    if (i < NN * NH)   mraw[i] = 0.0f;      // max(0, segment_max) starts at 0
    if (i < NN)        deg[i]  = 0u;
}

// ---------------------------------------------------------------------------
// Kernel P: fold edge projection into attention:  v[h][k] = sum_d We[h*32+d][k]*a[h][64+d]
// One block of 256 threads; thread t -> (h = t/64, k = t%64)
// ---------------------------------------------------------------------------
__global__ void k_fold_v(const float* __restrict__ We, const float* __restrict__ a,
                         float* __restrict__ v) {
    int t = threadIdx.x;
    int h = t >> 6, k = t & 63;
    float acc = 0.0f;
    #pragma unroll
    for (int d = 0; d < HD; ++d)
        acc += We[(h * HD + d) * EDIM + k] * a[h * (3 * HD) + 2 * HD + d];
    v[h * EDIM + k] = acc;
}

// ---------------------------------------------------------------------------
// Kernel A: h = X @ Wn^T  via V_WMMA_F32_16X16X4_F32
// Block = 256 threads = 8 waves; each wave owns one 16x16 output tile.
// blockIdx.x -> M strip (16 nodes), wave -> N tile (16 of the 128 outputs).
// A frag (16x4 f32): lanes 0-15 hold K={0,1}, lanes 16-31 hold K={2,3} (float2).
// B frag (4x16 f32): mirrored K split; both map to contiguous float2 loads here.
// C/D (16x16 f32): VGPR g, lane l -> row g + 8*(l>=16), col l%16.
// ---------------------------------------------------------------------------
__global__ __launch_bounds__(256) void k_node_gemm(const float* __restrict__ X,
                                                   const float* __restrict__ Wn,
                                                   float* __restrict__ hbuf) {
    const int lane = threadIdx.x & 31;
    const int wave = threadIdx.x >> 5;          // 0..7 -> N tile
    const int lm   = lane & 15;
    const int half = lane >> 4;                 // 0 or 1
    const int m0   = blockIdx.x * 16;
    const int n0   = wave * 16;

    int mrow = m0 + lm; if (mrow >= NN) mrow = NN - 1;   // clamp (NN is a mult of 16 anyway)
    const float* arow = X  + (size_t)mrow * NDIM + 2 * half;       // A[m, k + {0,1}]
    const float* brow = Wn + (size_t)(n0 + lm) * NDIM + 2 * half;  // Wn[n, k + {0,1}] = B[k,n]

    v8f c = {};
    #pragma unroll
    for (int k = 0; k < NDIM; k += 4) {
        v2f af = *(const v2f*)(arow + k);
        v2f bf = *(const v2f*)(brow + k);
        c = __builtin_amdgcn_wmma_f32_16x16x4_f32(
                /*neg_a=*/false, af, /*neg_b=*/false, bf,
                /*c_mod=*/(short)0, c, /*reuse_a=*/false, /*reuse_b=*/false);
    }

    #pragma unroll
    for (int g = 0; g < 8; ++g) {
        int m = m0 + g + 8 * half;
        if (m < NN) hbuf[(size_t)m * ODIM + n0 + lm] = c[g];
    }
}

// ---------------------------------------------------------------------------
// Kernel B: per-node attention scalars
//   s1[n,h] = h[n,h,:] . a[h,0:32]      (source term)
//   s2[n,h] = h[n,h,:] . a[h,32:64]     (target term)
// ---------------------------------------------------------------------------
__global__ void k_node_scalars(const float* __restrict__ hbuf, const float* __restrict__ a,
                               float* __restrict__ s1, float* __restrict__ s2) {
    int idx = blockIdx.x * blockDim.x + threadIdx.x;  // n*NH + h
    if (idx >= NN * NH) return;
    int n = idx >> 2, h = idx & 3;
    const float* hr = hbuf + (size_t)n * ODIM + h * HD;
    const float* ar = a + h * (3 * HD);
    float acc1 = 0.0f, acc2 = 0.0f;
    #pragma unroll
    for (int d = 0; d < HD; ++d) {
        float x = hr[d];
        acc1 += x * ar[d];
        acc2 += x * ar[HD + d];
    }
    s1[idx] = acc1;
    s2[idx] = acc2;
}

// ---------------------------------------------------------------------------
// Kernel C: per-edge logits + LeakyReLU; clamped atomic max -> mraw; deg count.
// att[e,h] = leaky( s1[src,h] + s2[tgt,h] + Ef[e] . v[h] )
// segment-max with uint atomicMax on max(att,0) bits (valid: values >= 0).
// ---------------------------------------------------------------------------
__global__ __launch_bounds__(256) void k_edge_att(const float* __restrict__ Ef,
                                                  const int* __restrict__ eidx,
                                                  const float* __restrict__ s1,
                                                  const float* __restrict__ s2,
                                                  const float* __restrict__ v,
                                                  float* __restrict__ att,
                                                  float* __restrict__ mraw,
                                                  unsigned* __restrict__ deg) {
    __shared__ float vs[NH * EDIM];
    if (threadIdx.x < NH * EDIM) vs[threadIdx.x] = v[threadIdx.x];
    __syncthreads();

    int e = blockIdx.x * blockDim.x + threadIdx.x;
    if (e >= NE) return;

    const float* row = Ef + (size_t)e * EDIM;
    float acc[NH] = {0.f, 0.f, 0.f, 0.f};
    #pragma unroll
    for (int k = 0; k < EDIM; k += 4) {
        float4 x = *(const float4*)(row + k);
        #pragma unroll
        for (int h = 0; h < NH; ++h) {
            const float4 vv = *(const float4*)(&vs[h * EDIM + k]);
            acc[h] += x.x * vv.x + x.y * vv.y + x.z * vv.z + x.w * vv.w;
        }
    }

    int s = eidx[e];        // edge_indices[0, e]
    int t = eidx[NE + e];   // edge_indices[1, e]
    #pragma unroll
    for (int h = 0; h < NH; ++h) {
        float av = s1[s * NH + h] + s2[t * NH + h] + acc[h];
        av = av > 0.0f ? av : 0.2f * av;          // LeakyReLU(0.2)
        att[(size_t)e * NH + h] = av;
        float amax = fmaxf(av, 0.0f);             // m = max(0, seg_max) anyway
        atomicMax((unsigned*)&mraw[t * NH + h], __float_as_uint(amax));
    }
    atomicAdd(&deg[t], 1u);
}

// ---------------------------------------------------------------------------
// Kernel D: denominator base: denom[n,h] = (E - deg[n]) * exp(-m[n,h])
// ---------------------------------------------------------------------------
__global__ void k_denom_base(const float* __restrict__ mraw, const unsigned* __restrict__ deg,
                             float* __restrict__ denom) {
    int idx = blockIdx.x * blockDim.x + threadIdx.x;
    if (idx >= NN * NH) return;
    int n = idx >> 2;
    float m = mraw[idx];
    denom[idx] = (float)(NE - deg[n]) * __expf(-m);
}

// ---------------------------------------------------------------------------
// Kernel E: ex = exp(att - m[tgt]); accumulate into denom; overwrite att with ex.
// ---------------------------------------------------------------------------
__global__ __launch_bounds__(256) void k_edge_exp(const int* __restrict__ eidx,
                                                  const float* __restrict__ mraw,
                                                  float* __restrict__ att,
                                                  float* __restrict__ denom) {
    int e = blockIdx.x * blockDim.x + threadIdx.x;
    if (e >= NE) return;
    int t = eidx[NE + e];
    #pragma unroll
    for (int h = 0; h < NH; ++h) {
        float ex = __expf(att[(size_t)e * NH + h] - mraw[t * NH + h]);
        att[(size_t)e * NH + h] = ex;
        atomicAdd(&denom[t * NH + h], ex);
    }
}

// ---------------------------------------------------------------------------
// Kernel F: weighted scatter-add.  One wave per edge, lane -> 4 output floats.
// out[tgt, h*32+d] += (ex/denom[tgt,h]) * h[src, h*32+d]
// ---------------------------------------------------------------------------
__global__ __launch_bounds__(256) void k_aggregate(const float* __restrict__ hbuf,
                                                   const int* __restrict__ eidx,
                                                   const float* __restrict__ ex,
                                                   const float* __restrict__ denom,
                                                   float* __restrict__ out) {
    int lane = threadIdx.x & 31;
    int e = blockIdx.x * 8 + (threadIdx.x >> 5);
    if (e >= NE) return;
    int s = eidx[e];
    int t = eidx[NE + e];
    int h = lane >> 3;                         // (lane*4)/32
    float w = ex[(size_t)e * NH + h] / denom[t * NH + h];
    float4 hv = *(const float4*)(hbuf + (size_t)s * ODIM + lane * 4);
    float* o = out + (size_t)t * ODIM + lane * 4;
    atomicAdd(o + 0, w * hv.x);
    atomicAdd(o + 1, w * hv.y);
    atomicAdd(o + 2, w * hv.z);
    atomicAdd(o + 3, w * hv.w);
}

// ---------------------------------------------------------------------------
// Host launcher
// ---------------------------------------------------------------------------
extern "C" void kernel_launch(void* const* d_in, const int* in_sizes, int n_in,
                              void* d_out, int out_size, void* d_ws, size_t ws_size,
                              hipStream_t stream) {
    const float* X  = (const float*)d_in[0];   // [NN, 128]
    const int*   EI = (const int*)  d_in[1];   // [2, NE]
    const float* Ef = (const float*)d_in[2];   // [NE, 64]
    const float* Wn = (const float*)d_in[3];   // [128, 128]
    const float* We = (const float*)d_in[4];   // [128, 64]
    const float* a  = (const float*)d_in[5];   // [4, 96]
    float* out = (float*)d_out;                // [NN, 128]

    // Workspace partition (byte offsets, 256-aligned)
    char* ws = (char*)d_ws;
    size_t off = 0;
    auto alloc = [&](size_t bytes) { char* p = ws + off; off = (off + bytes + 255) & ~(size_t)255; return p; };
    float*    hbuf  = (float*)   alloc((size_t)NN * ODIM * 4);   // 25.6 MB
    float*    s1    = (float*)   alloc((size_t)NN * NH * 4);
    float*    s2    = (float*)   alloc((size_t)NN * NH * 4);
    float*    mraw  = (float*)   alloc((size_t)NN * NH * 4);
    unsigned* deg   = (unsigned*)alloc((size_t)NN * 4);
    float*    denom = (float*)   alloc((size_t)NN * NH * 4);
    float*    attb  = (float*)   alloc((size_t)NE * NH * 4);     // 9.6 MB (att, then ex)
    float*    vfold = (float*)   alloc((size_t)NH * EDIM * 4);
    (void)ws_size; (void)in_sizes; (void)n_in; (void)out_size;

    const int T = 256;
    k_init        <<<(NN * ODIM + T - 1) / T, T, 0, stream>>>(out, mraw, deg);
    k_fold_v      <<<1, T, 0, stream>>>(We, a, vfold);
    k_node_gemm   <<<(NN + 15) / 16, T, 0, stream>>>(X, Wn, hbuf);
    k_node_scalars<<<(NN * NH + T - 1) / T, T, 0, stream>>>(hbuf, a, s1, s2);
    k_edge_att    <<<(NE + T - 1) / T, T, 0, stream>>>(Ef, EI, s1, s2, vfold, attb, mraw, deg);
    k_denom_base  <<<(NN * NH + T - 1) / T, T, 0, stream>>>(mraw, deg, denom);
    k_edge_exp    <<<(NE + T - 1) / T, T, 0, stream>>>(EI, mraw, attb, denom);
    k_aggregate   <<<(NE + 7) / 8, T, 0, stream>>>(hbuf, EI, attb, denom, out);
}